// HunYuanMoEGate_56650618635036
// MI455X (gfx1250) — compile-verified
//
#include <hip/hip_runtime.h>

// MoE gate for MI455X (gfx1250, wave32).
// logits = H[32768,4096] @ W[16,4096]^T in fp32 via V_WMMA_F32_16X16X4_F32,
// softmax+top2+renorm folded into: w1 = 1/(1+exp(l2-l1)), w2 = 1-w1.
//
// HBM-bound: H streamed once (512 MB -> ~22us at 23.3 TB/s). W (256 KB)
// parked in LDS. One wave = one 16-token x 16-expert tile.
// Two independent WMMA accumulator chains break the D->C serial dependency
// (removes the per-WMMA s_delay_alu stalls seen with a single chain).

typedef __attribute__((ext_vector_type(2))) float v2f;
typedef __attribute__((ext_vector_type(4))) float v4f;
typedef __attribute__((ext_vector_type(8))) float v8f;

#define TOKENS  32768
#define DIM     4096
#define NEXP    16
#define WAVES   8            // 256 threads / wave32
#define BLOCKS  128
#define NTILES  (TOKENS / 16)
#define CPAD    17           // padded row stride for epilogue scratch

// One K=8 fp32 WMMA step accumulating a float4-pair (two 16x16x4 ops).
// K-slot convention: slots {0,1} come from lanes 0-15, slots {2,3} from
// lanes 16-31 (per ISA A-matrix 16x4 layout). A and B are loaded with the
// identical per-lane K addressing, so the per-instruction K grouping
// {k0, k0+1, k0+16, k0+17} is consistent between A and B.
__device__ __forceinline__ v8f wmma_acc_f32(v4f a, v4f b, v8f c) {
  v2f alo = __builtin_shufflevector(a, a, 0, 1);
  v2f ahi = __builtin_shufflevector(a, a, 2, 3);
  v2f blo = __builtin_shufflevector(b, b, 0, 1);
  v2f bhi = __builtin_shufflevector(b, b, 2, 3);
  // 8 args: (neg_a, A, neg_b, B, c_mod, C, reuse_a, reuse_b)
  c = __builtin_amdgcn_wmma_f32_16x16x4_f32(false, alo, false, blo, (short)0, c, false, false);
  c = __builtin_amdgcn_wmma_f32_16x16x4_f32(false, ahi, false, bhi, (short)0, c, false, false);
  return c;
}

__global__ __launch_bounds__(256) void moe_gate_wmma(
    const float* __restrict__ H,      // [TOKENS, DIM]
    const float* __restrict__ W,      // [NEXP, DIM]
    int*   __restrict__ idx_out,      // [TOKENS, 2]
    float* __restrict__ w_out)        // [TOKENS, 2]
{
  extern __shared__ float smem[];
  float* W_lds = smem;                              // NEXP*DIM floats (256 KB)

  const int tid  = threadIdx.x;
  const int wave = tid >> 5;
  const int lane = tid & 31;

  // ---- cooperative W -> LDS fill, coalesced b128 (layout identical to HBM) --
  {
    const v4f* Wv = (const v4f*)W;
    v4f*       Sv = (v4f*)W_lds;
    for (int i = tid; i < NEXP * DIM / 4; i += 256) Sv[i] = Wv[i];
  }
  __syncthreads();

  float* c_scr = smem + NEXP * DIM + wave * (16 * CPAD); // per-wave epilogue tile

  const int sub  = lane & 15;   // token-within-tile (A) / expert column (B)
  const int half = lane >> 4;   // K sub-block select
  const float* wrow = W_lds + sub * DIM + 16 * half;

  for (int tile = blockIdx.x * WAVES + wave; tile < NTILES;
       tile += gridDim.x * WAVES) {
    const float* hrow = H + (size_t)(tile * 16 + sub) * DIM + 16 * half;

    v8f c0 = {};   // even-chunk accumulator chain
    v8f c1 = {};   // odd-chunk accumulator chain (independent -> no WMMA stalls)
    // 32 K-values per iteration: lanes l and l+16 together consume one full
    // 128B cacheline of their row. 4 global b128 + 4 ds b128 + 8 WMMA,
    // alternating between the two accumulator chains.
    for (int k0 = 0; k0 < DIM; k0 += 32) {
      v4f a0 = *(const v4f*)(hrow + k0);
      v4f a1 = *(const v4f*)(hrow + k0 + 4);
      v4f a2 = *(const v4f*)(hrow + k0 + 8);
      v4f a3 = *(const v4f*)(hrow + k0 + 12);
      v4f b0 = *(const v4f*)(wrow + k0);
      v4f b1 = *(const v4f*)(wrow + k0 + 4);
      v4f b2 = *(const v4f*)(wrow + k0 + 8);
      v4f b3 = *(const v4f*)(wrow + k0 + 12);
      c0 = wmma_acc_f32(a0, b0, c0);
      c1 = wmma_acc_f32(a1, b1, c1);
      c0 = wmma_acc_f32(a2, b2, c0);
      c1 = wmma_acc_f32(a3, b3, c1);
    }
    v8f c = c0 + c1;  // merge chains (8 VALU adds per tile)

    // ---- epilogue: transpose tile through LDS, per-token top-2 --------------
    // C layout: VGPR j, lanes 0-15 -> M=j, lanes 16-31 -> M=j+8; N=lane&15.
    const int mbase = half * 8;
#pragma unroll
    for (int j = 0; j < 8; ++j)
      c_scr[(mbase + j) * CPAD + sub] = c[j];

    asm volatile("s_wait_dscnt 0" ::: "memory");

    if (lane < 16) {
      const float* row = c_scr + lane * CPAD;
      float best = row[0];   int bi = 0;
      float second = -3.402823466e+38f; int si = 0;
#pragma unroll
      for (int e = 1; e < NEXP; ++e) {
        float v = row[e];
        if (v > best)        { second = best; si = bi; best = v; bi = e; }
        else if (v > second) { second = v; si = e; }
      }
      // softmax + top2 + renorm collapses to a single sigmoid of (l2-l1)
      float r   = __expf(second - best);
      float inv = 1.0f / (1.0f + r);
      int token = tile * 16 + lane;
      idx_out[2 * token]     = bi;
      idx_out[2 * token + 1] = si;
      w_out[2 * token]       = inv;
      w_out[2 * token + 1]   = r * inv;
    }
  }
}

extern "C" void kernel_launch(void* const* d_in, const int* in_sizes, int n_in,
                              void* d_out, int out_size, void* d_ws, size_t ws_size,
                              hipStream_t stream) {
  const float* H = (const float*)d_in[0];   // [32768, 4096] fp32
  const float* W = (const float*)d_in[1];   // [16, 4096] fp32

  // d_out = concat(topk_indices int32 [T,2], topk_weights fp32 [T,2])
  int*   idx_out = (int*)d_out;
  float* w_out   = (float*)d_out + (size_t)TOKENS * 2;

  size_t shmem = (size_t)(NEXP * DIM) * sizeof(float)        // W tile
               + (size_t)(WAVES * 16 * CPAD) * sizeof(float); // epilogue scratch
  hipLaunchKernelGGL(moe_gate_wmma, dim3(BLOCKS), dim3(256), shmem, stream,
                     H, W, idx_out, w_out);
}